// DeterministicMaxPool3d_79740362818108
// MI455X (gfx1250) — compile-verified
//
#include <hip/hip_runtime.h>
#include <math.h>

// MaxPool3d: kernel=3, stride=2, pad=1 (-inf), input (2,32,128,128,128) f32.
// Output (2,32,64,64,64) f32.
//
// Memory-bound: ~576 MB traffic / 23.3 TB/s ~= 25 us floor. Strategy:
// stage a 9x9x128 f32 input tile (41.5 KB) into LDS with gfx1250 async
// global->LDS copies (ASYNCcnt), then each thread computes 4 outputs from
// LDS with bounds-checked 3x3x3 max.

#define IN_D   128
#define OUT_D  64
#define TD     4            // od tile
#define THH    4            // oh tile
#define LDROWS 9            // 2*TD+1 input d-rows
#define LHROWS 9            // 2*THH+1 input h-rows
#define LW     128          // full w extent staged

typedef int v4i __attribute__((ext_vector_type(4)));
typedef __attribute__((address_space(1))) v4i* g_v4i_p;   // global int4*
typedef __attribute__((address_space(3))) v4i* l_v4i_p;   // LDS int4*

__device__ __forceinline__ void async_copy16(const float* g, float* l) {
#if __has_builtin(__builtin_amdgcn_global_load_async_to_lds_b128)
  __builtin_amdgcn_global_load_async_to_lds_b128(
      (g_v4i_p)const_cast<float*>(g), (l_v4i_p)l, /*offset=*/0, /*cpol=*/0);
#else
  *(float4*)l = *(const float4*)g;   // fallback: VGPR-staged copy
#endif
}

__device__ __forceinline__ void wait_async_zero() {
#if __has_builtin(__builtin_amdgcn_s_wait_asynccnt)
  __builtin_amdgcn_s_wait_asynccnt(0);
#else
  asm volatile("s_wait_asynccnt 0" ::: "memory");
#endif
}

__global__ __launch_bounds__(256)
void maxpool3d_k3s2p1_kernel(const float* __restrict__ x,
                             float* __restrict__ out) {
  __shared__ float tile[LDROWS * LHROWS * LW];  // 41472 bytes

  const int tid   = threadIdx.x;
  const int plane = blockIdx.z;                 // n*32 + c, 0..63
  const int od0   = blockIdx.y * TD;            // output-d tile origin
  const int oh0   = blockIdx.x * THH;           // output-h tile origin
  const int dbase = 2 * od0 - 1;                // input-d of tile row 0
  const int hbase = 2 * oh0 - 1;                // input-h of tile row 0

  const float* __restrict__ src =
      x + (size_t)plane * IN_D * IN_D * IN_D;

  // ---- Stage input tile into LDS: 9*9 rows x 128 floats, 16B chunks ----
  const int nchunk = LDROWS * LHROWS * (LW / 4);      // 2592
  for (int c = tid; c < nchunk; c += 256) {
    const int ld  = c / (LHROWS * (LW / 4));
    const int r   = c % (LHROWS * (LW / 4));
    const int lh  = r / (LW / 4);
    const int lw4 = r % (LW / 4);
    const int d = dbase + ld;
    const int h = hbase + lh;
    if ((unsigned)d < (unsigned)IN_D && (unsigned)h < (unsigned)IN_D) {
      const float* g = src + ((size_t)d * IN_D + h) * IN_D + lw4 * 4;
      float*       l = &tile[(ld * LHROWS + lh) * LW + lw4 * 4];
      async_copy16(g, l);
    }
  }
  wait_async_zero();     // my async copies landed in LDS
  __syncthreads();       // everyone else's too

  // ---- Compute: each thread -> 4 outputs (same ow, 4 of 16 (od,oh)) ----
  const int ow = tid & 63;
  const int wb = 2 * ow - 1;

  for (int p = tid >> 6; p < TD * THH; p += 4) {
    const int odl = p >> 2;
    const int ohl = p & 3;
    float m = -INFINITY;
#pragma unroll
    for (int kd = 0; kd < 3; ++kd) {
      const int d = dbase + 2 * odl + kd;      // global input d
      if ((unsigned)d >= (unsigned)IN_D) continue;
      const int ld = 2 * odl + kd;             // tile row
#pragma unroll
      for (int kh = 0; kh < 3; ++kh) {
        const int h = hbase + 2 * ohl + kh;
        if ((unsigned)h >= (unsigned)IN_D) continue;
        const int lh = 2 * ohl + kh;
        const float* __restrict__ row = &tile[(ld * LHROWS + lh) * LW];
#pragma unroll
        for (int kw = 0; kw < 3; ++kw) {
          const int w = wb + kw;
          if ((unsigned)w < (unsigned)IN_D) m = fmaxf(m, row[w]);
        }
      }
    }
    const int od = od0 + odl;
    const int oh = oh0 + ohl;
    out[(((size_t)plane * OUT_D + od) * OUT_D + oh) * OUT_D + ow] = m;
  }
}

extern "C" void kernel_launch(void* const* d_in, const int* in_sizes, int n_in,
                              void* d_out, int out_size, void* d_ws, size_t ws_size,
                              hipStream_t stream) {
  const float* x = (const float*)d_in[0];
  float* out     = (float*)d_out;
  // grid: (oh tiles, od tiles, planes) = (16, 16, 64); 256 threads = 8 wave32
  dim3 grid(OUT_D / THH, OUT_D / TD, 2 * 32);
  maxpool3d_k3s2p1_kernel<<<grid, 256, 0, stream>>>(x, out);
}